// CausalSelfAttention_34626026341018
// MI455X (gfx1250) — compile-verified
//
#include <hip/hip_runtime.h>
#include <hip/hip_bf16.h>

// MLA attention (fp32) for MI455X / gfx1250.
// All matrix math via V_WMMA_F32_16X16X4_F32 (fp32 in/out, matches reference
// numerics). Flash-style fused attention: no [B,H,T,T] materialization.
// Round 2: software-pipelined fragment loads (double-buffered chunks in the
// GEMM, batched fragment arrays in attention) so WMMAs don't serialize on
// per-load s_wait_loadcnt 0.

typedef __attribute__((ext_vector_type(2))) float v2f;
typedef __attribute__((ext_vector_type(8))) float v8f;

#define T_SEQ 2048
#define C_DIM 1024
#define N_HEAD 16
#define HEAD_DIM 64
#define KV_DIM 64
#define QL_DIM 512

__device__ __forceinline__ v8f wmma_f32(v2f a, v2f b, v8f c) {
  // D = A(16x4,f32) * B(4x16,f32) + C(16x16,f32)
  return __builtin_amdgcn_wmma_f32_16x16x4_f32(
      /*neg_a=*/false, a, /*neg_b=*/false, b,
      /*c_mod=*/(short)0, c, /*reuse_a=*/false, /*reuse_b=*/false);
}

// ---------------------------------------------------------------------------
// Generic GEMM: Y[M,N] = X[M,K] @ W[N,K]^T + bias[N]
// One wave computes one 32x64 output tile (2 M-frags x 4 N-frags = 8 accs).
// K is consumed in chunks of 16 (4 WMMA k-steps), double-buffered so the next
// chunk's 24 b64 loads are in flight while the current 32 WMMAs execute.
// Requires K % 32 == 0 (true for 64 / 512 / 1024).
// ---------------------------------------------------------------------------
struct Chunk {
  v2f a0[4];     // A rows 0..15 of tile, k-steps 0..3
  v2f a1[4];     // A rows 16..31
  v2f b[4][4];   // [k-step][n-frag]
};

__global__ __launch_bounds__(256)
void wmma_gemm_xwT(const float* __restrict__ X, const float* __restrict__ W,
                   const float* __restrict__ bias, float* __restrict__ Y,
                   int M, int N, int K) {
  const int lane  = threadIdx.x & 31;
  const int wave  = threadIdx.x >> 5;
  const int tile  = blockIdx.x * (blockDim.x >> 5) + wave;
  const int ntn   = N >> 6;                 // N-tiles of width 64
  const int mt    = tile / ntn;
  const int nt    = tile % ntn;
  if (mt >= (M >> 5)) return;

  const int r     = lane & 15;
  const int khalf = (lane >> 4) * 2;        // lane-half K offset (0 or 2)
  const int n0    = nt * 64;

  const float* __restrict__ x0 = X + (size_t)(mt * 32 + r) * K;
  const float* __restrict__ x1 = x0 + (size_t)16 * K;
  const float* __restrict__ w0 = W + (size_t)(n0 + r) * K;

  v8f acc[8] = {};
  Chunk c0, c1;

  auto load_chunk = [&](Chunk& c, int kbase) {
#pragma unroll
    for (int i = 0; i < 4; ++i) {
      const int kk = kbase + i * 4 + khalf;
      c.a0[i].x = x0[kk]; c.a0[i].y = x0[kk + 1];
      c.a1[i].x = x1[kk]; c.a1[i].y = x1[kk + 1];
#pragma unroll
      for (int j = 0; j < 4; ++j) {
        const float* wr = w0 + (size_t)(j * 16) * K;
        c.b[i][j].x = wr[kk]; c.b[i][j].y = wr[kk + 1];
      }
    }
  };
  auto compute = [&](const Chunk& c) {
#pragma unroll
    for (int i = 0; i < 4; ++i) {
#pragma unroll
      for (int j = 0; j < 4; ++j) {
        acc[j]     = wmma_f32(c.a0[i], c.b[i][j], acc[j]);
        acc[4 + j] = wmma_f32(c.a1[i], c.b[i][j], acc[4 + j]);
      }
    }
  };

  load_chunk(c0, 0);
  for (int k0 = 0; k0 < K; k0 += 32) {
    load_chunk(c1, k0 + 16);                 // prefetch while computing c0
    compute(c0);
    if (k0 + 32 < K) load_chunk(c0, k0 + 32);
    compute(c1);
  }

  // D layout: VGPR i -> row (i + 8*(lane>>4)), col = lane&15
  const int rbase = mt * 32 + 8 * (lane >> 4);
#pragma unroll
  for (int j = 0; j < 4; ++j) {
    const int col = n0 + j * 16 + r;
    const float bc = bias ? bias[col] : 0.0f;
#pragma unroll
    for (int i = 0; i < 8; ++i) {
      Y[(size_t)(rbase + i) * N + col]      = acc[j][i] + bc;
      Y[(size_t)(rbase + 16 + i) * N + col] = acc[4 + j][i] + bc;
    }
  }
}

// ---------------------------------------------------------------------------
// Fused flash attention (no causal mask, per reference).
// One wave per (b, h, 16-query-row block). Shared latent K across heads (L2
// catches the 16x reuse). Online softmax; P transposed via per-wave LDS.
// K/V fragments for each key block are loaded as batched arrays so the loads
// pipeline ahead of the WMMAs and softmax VALU work.
// ---------------------------------------------------------------------------
__global__ __launch_bounds__(256)
void mla_flash_attn(const float* __restrict__ Qb,  // [B,T,H*KV]
                    const float* __restrict__ Kb,  // [B,T,KV]   (c_kv)
                    const float* __restrict__ Vb,  // [B,T,H*HD]
                    float* __restrict__ Yb,        // [B,T,H*HD]
                    int Bn) {
  __shared__ float lds[8][16 * 20];                // stride 20: bank-safe
  const int lane = threadIdx.x & 31;
  const int wave = threadIdx.x >> 5;
  float* pbuf = lds[wave];

  const int gw      = blockIdx.x * 8 + wave;
  const int qblocks = T_SEQ / 16;
  const int bh      = gw / qblocks;
  const int qb      = gw % qblocks;
  if (bh >= Bn * N_HEAD) return;
  const int b = bh / N_HEAD, h = bh % N_HEAD;

  const int r     = lane & 15;
  const int khalf = (lane >> 4) * 2;
  const float scale = 0.125f;                      // 1/sqrt(64)

  // Resident Q block: 16 rows x 64 dims -> 16 A-fragments.
  const int q0 = qb * 16;
  const float* qrow =
      Qb + ((size_t)b * T_SEQ + q0 + r) * (N_HEAD * KV_DIM) + h * KV_DIM;
  v2f qf[16];
#pragma unroll
  for (int d = 0; d < 16; ++d) {
    qf[d].x = qrow[d * 4 + khalf];
    qf[d].y = qrow[d * 4 + khalf + 1];
  }

  v8f o[4] = {};
  float m[8], l[8];
#pragma unroll
  for (int j = 0; j < 8; ++j) { m[j] = -1.0e30f; l[j] = 0.0f; }

  const float* kbase = Kb + (size_t)b * T_SEQ * KV_DIM;
  const float* vbase = Vb + (size_t)b * T_SEQ * (N_HEAD * HEAD_DIM) +
                       h * HEAD_DIM;

  for (int s0 = 0; s0 < T_SEQ; s0 += 16) {
    // ---- batched fragment loads for this key block ----
    const float* krow = kbase + (size_t)(s0 + r) * KV_DIM;
    v2f kf[16];
#pragma unroll
    for (int d = 0; d < 16; ++d) {
      kf[d].x = krow[d * 4 + khalf];
      kf[d].y = krow[d * 4 + khalf + 1];
    }
    v2f vf[4][4];                                   // [k-step][n-frag]
#pragma unroll
    for (int kk = 0; kk < 4; ++kk) {
      const float* vrow =
          vbase + (size_t)(s0 + kk * 4 + khalf) * (N_HEAD * HEAD_DIM);
#pragma unroll
      for (int n = 0; n < 4; ++n) {
        vf[kk][n].x = vrow[n * 16 + r];
        vf[kk][n].y = vrow[(N_HEAD * HEAD_DIM) + n * 16 + r];
      }
    }

    // ---- S = Q @ K^T (16x16 tile, 16 WMMA k-steps) ----
    v8f s = {};
#pragma unroll
    for (int d = 0; d < 16; ++d) s = wmma_f32(qf[d], kf[d], s);

    // ---- online softmax: row j+8*(lane>>4) lives in s[j] across 16 lanes ----
    float p[8], alpha[8];
#pragma unroll
    for (int j = 0; j < 8; ++j) {
      float sj = s[j] * scale;
      float mx = sj;
      mx = fmaxf(mx, __shfl_xor(mx, 1));
      mx = fmaxf(mx, __shfl_xor(mx, 2));
      mx = fmaxf(mx, __shfl_xor(mx, 4));
      mx = fmaxf(mx, __shfl_xor(mx, 8));           // reduce within lane-half
      float mnew = fmaxf(m[j], mx);
      float pj = __expf(sj - mnew);
      float al = __expf(m[j] - mnew);
      float rs = pj;
      rs += __shfl_xor(rs, 1);
      rs += __shfl_xor(rs, 2);
      rs += __shfl_xor(rs, 4);
      rs += __shfl_xor(rs, 8);
      l[j] = l[j] * al + rs;
      m[j] = mnew;
      alpha[j] = al;
      p[j] = pj;
    }
#pragma unroll
    for (int n = 0; n < 4; ++n)
#pragma unroll
      for (int j = 0; j < 8; ++j) o[n][j] *= alpha[j];

    // ---- transpose P: D-layout -> LDS -> A-layout (same-wave, DS in-order) --
#pragma unroll
    for (int j = 0; j < 8; ++j)
      pbuf[(j + 8 * (lane >> 4)) * 20 + r] = p[j];

    // ---- O += P @ V ----
#pragma unroll
    for (int kk = 0; kk < 4; ++kk) {
      v2f pa;
      pa.x = pbuf[r * 20 + kk * 4 + khalf];
      pa.y = pbuf[r * 20 + kk * 4 + khalf + 1];
#pragma unroll
      for (int n = 0; n < 4; ++n) o[n] = wmma_f32(pa, vf[kk][n], o[n]);
    }
  }

  // ---- normalize rows and store ----
#pragma unroll
  for (int j = 0; j < 8; ++j) {
    const float inv = 1.0f / l[j];
    const int row = q0 + j + 8 * (lane >> 4);
    float* yrow =
        Yb + ((size_t)b * T_SEQ + row) * (N_HEAD * HEAD_DIM) + h * HEAD_DIM;
#pragma unroll
    for (int n = 0; n < 4; ++n) yrow[n * 16 + r] = o[n][j] * inv;
  }
}

// ---------------------------------------------------------------------------
extern "C" void kernel_launch(void* const* d_in, const int* in_sizes, int n_in,
                              void* d_out, int out_size, void* d_ws,
                              size_t ws_size, hipStream_t stream) {
  const float* x      = (const float*)d_in[0];
  const float* wq_d   = (const float*)d_in[1];
  const float* bq_d   = (const float*)d_in[2];
  const float* w_qk   = (const float*)d_in[3];
  const float* b_qk   = (const float*)d_in[4];
  const float* wkv_d  = (const float*)d_in[5];
  const float* bkv_d  = (const float*)d_in[6];
  const float* wv_u   = (const float*)d_in[7];
  const float* bv_u   = (const float*)d_in[8];
  const float* w_proj = (const float*)d_in[9];
  const float* b_proj = (const float*)d_in[10];
  float* out = (float*)d_out;

  const int Bn = in_sizes[0] / (T_SEQ * C_DIM);   // = 4
  const int M  = Bn * T_SEQ;                      // = 8192

  float* cq   = (float*)d_ws;
  float* ckv  = cq   + (size_t)M * QL_DIM;
  float* qbuf = ckv  + (size_t)M * KV_DIM;
  float* vbuf = qbuf + (size_t)M * (N_HEAD * KV_DIM);
  float* ybuf = vbuf + (size_t)M * (N_HEAD * HEAD_DIM);

  auto gemm = [&](const float* X, const float* W, const float* bias, float* Y,
                  int n, int k) {
    const int tiles  = (M / 32) * (n / 64);
    const int blocks = (tiles + 7) / 8;             // 8 waves / 256-thr block
    hipLaunchKernelGGL(wmma_gemm_xwT, dim3(blocks), dim3(256), 0, stream,
                       X, W, bias, Y, M, n, k);
  };

  gemm(x,    wq_d,  bq_d,  cq,   QL_DIM,            C_DIM);   // c_q
  gemm(x,    wkv_d, bkv_d, ckv,  KV_DIM,            C_DIM);   // c_kv
  gemm(cq,   w_qk,  b_qk,  qbuf, N_HEAD * KV_DIM,   QL_DIM);  // q
  gemm(ckv,  wv_u,  bv_u,  vbuf, N_HEAD * HEAD_DIM, KV_DIM);  // v

  const int awaves  = Bn * N_HEAD * (T_SEQ / 16);   // 8192
  const int ablocks = (awaves + 7) / 8;             // 1024
  hipLaunchKernelGGL(mla_flash_attn, dim3(ablocks), dim3(256), 0, stream,
                     qbuf, ckv, vbuf, ybuf, Bn);

  gemm(ybuf, w_proj, b_proj, out, C_DIM, C_DIM);              // out proj
}